// Loss_HF_86079734546730
// MI455X (gfx1250) — compile-verified
//
#include <hip/hip_runtime.h>
#include <math.h>

// ---------------------------------------------------------------------------
// Problem constants
// ---------------------------------------------------------------------------
#define BATCH      64
#define CHANS      3
#define IMG        512
#define HIMG       256                  // 256x256 per subband
#define PLANE_N    (HIMG * HIMG)        // 65536 elements per (b,c) subband
#define K_FEAT     (CHANS * PLANE_N)    // 196608 features per batch row
#define EPS_STD    1e-5f
#define EPS_COS    1e-8f
#define EPS_P      1e-8f

// Gram partitioning: 256 wave-chunks per matrix, each covering KSUB features
#define NWCHUNK    256
#define KSUB       (K_FEAT / NWCHUNK)   // 768, multiple of 4

typedef __attribute__((ext_vector_type(2))) float v2f;
typedef __attribute__((ext_vector_type(8))) float v8f;

// ---------------------------------------------------------------------------
// Kernel 1: Haar high-frequency + per-plane normalize.
// One block per (b,c) plane. Pass 1: stats of lh/hl/hh. Pass 2: write
// H[b][c*65536 + y*256 + x] = (nlh + nhl + nhh) / 3.
// ---------------------------------------------------------------------------
__global__ __launch_bounds__(256) void haar_norm_kernel(
    const float* __restrict__ Z, float* __restrict__ Hout)
{
    const int plane = blockIdx.x;            // 0..191
    const int b = plane / CHANS;
    const int c = plane % CHANS;
    const float* __restrict__ in = Z + (size_t)plane * (IMG * IMG);
    float* __restrict__ out = Hout + (size_t)b * K_FEAT + (size_t)c * PLANE_N;

    const int tid = threadIdx.x;
    float s[6] = {0.f, 0.f, 0.f, 0.f, 0.f, 0.f};   // sum/sumsq for lh,hl,hh

    // Pass 1: each thread handles pairs of Haar quads via float4 loads.
    for (int p = tid; p < PLANE_N / 2; p += 256) {
        const int y  = p >> 7;          // 0..255
        const int xp = p & 127;         // covers x = 2*xp, 2*xp+1
        const float4 r0 = *(const float4*)(in + (size_t)(2 * y) * IMG + 4 * xp);
        const float4 r1 = *(const float4*)(in + (size_t)(2 * y + 1) * IMG + 4 * xp);
        // quad 0: a=r0.x b=r0.y c=r1.x d=r1.y
        float lh0 = ( r0.x - r0.y + r1.x - r1.y) * 0.5f;
        float hl0 = ( r0.x + r0.y - r1.x - r1.y) * 0.5f;
        float hh0 = (-r0.x + r0.y + r1.x - r1.y) * 0.5f;
        // quad 1: a=r0.z b=r0.w c=r1.z d=r1.w
        float lh1 = ( r0.z - r0.w + r1.z - r1.w) * 0.5f;
        float hl1 = ( r0.z + r0.w - r1.z - r1.w) * 0.5f;
        float hh1 = (-r0.z + r0.w + r1.z - r1.w) * 0.5f;
        s[0] += lh0 + lh1;  s[1] += lh0 * lh0 + lh1 * lh1;
        s[2] += hl0 + hl1;  s[3] += hl0 * hl0 + hl1 * hl1;
        s[4] += hh0 + hh1;  s[5] += hh0 * hh0 + hh1 * hh1;
    }

    // wave32 reduction
    #pragma unroll
    for (int off = 16; off > 0; off >>= 1) {
        #pragma unroll
        for (int v = 0; v < 6; ++v) s[v] += __shfl_xor(s[v], off, 32);
    }
    __shared__ float wsum[8][6];
    __shared__ float stats[6];          // mean[3], inv_std[3]
    const int wave = tid >> 5;
    const int lane = tid & 31;
    if (lane == 0) {
        #pragma unroll
        for (int v = 0; v < 6; ++v) wsum[wave][v] = s[v];
    }
    __syncthreads();
    if (tid == 0) {
        float tot[6] = {0.f, 0.f, 0.f, 0.f, 0.f, 0.f};
        for (int w = 0; w < 8; ++w)
            #pragma unroll
            for (int v = 0; v < 6; ++v) tot[v] += wsum[w][v];
        #pragma unroll
        for (int sb = 0; sb < 3; ++sb) {
            float mean = tot[2 * sb] * (1.f / (float)PLANE_N);
            float var  = (tot[2 * sb + 1] - (float)PLANE_N * mean * mean)
                         * (1.f / (float)(PLANE_N - 1));   // ddof = 1
            var = fmaxf(var, 0.f);
            stats[sb]     = mean;
            stats[3 + sb] = 1.f / (sqrtf(var) + EPS_STD);
        }
    }
    __syncthreads();
    const float m0 = stats[0], m1 = stats[1], m2 = stats[2];
    const float i0 = stats[3], i1 = stats[4], i2 = stats[5];

    // Pass 2: recompute (plane is L2-hot) and write normalized average.
    for (int p = tid; p < PLANE_N / 2; p += 256) {
        const int y  = p >> 7;
        const int xp = p & 127;
        const float4 r0 = *(const float4*)(in + (size_t)(2 * y) * IMG + 4 * xp);
        const float4 r1 = *(const float4*)(in + (size_t)(2 * y + 1) * IMG + 4 * xp);
        float lh0 = ( r0.x - r0.y + r1.x - r1.y) * 0.5f;
        float hl0 = ( r0.x + r0.y - r1.x - r1.y) * 0.5f;
        float hh0 = (-r0.x + r0.y + r1.x - r1.y) * 0.5f;
        float lh1 = ( r0.z - r0.w + r1.z - r1.w) * 0.5f;
        float hl1 = ( r0.z + r0.w - r1.z - r1.w) * 0.5f;
        float hh1 = (-r0.z + r0.w + r1.z - r1.w) * 0.5f;
        float2 h;
        h.x = ((lh0 - m0) * i0 + (hl0 - m1) * i1 + (hh0 - m2) * i2) * (1.f / 3.f);
        h.y = ((lh1 - m0) * i0 + (hl1 - m1) * i1 + (hh1 - m2) * i2) * (1.f / 3.f);
        *(float2*)(out + (size_t)y * HIMG + 2 * xp) = h;
    }
}

// ---------------------------------------------------------------------------
// Kernel 2: 64x64 Gram matrix G = H * H^T via V_WMMA_F32_16X16X4_F32.
// With the fp32 WMMA layouts, the A-fragment of row-tile t and the
// B-fragment of col-tile t are the SAME v2f load:
//   frag[t](lane) = { X[t*16 + (lane&15)][k + 2*(lane>>4)],
//                     X[t*16 + (lane&15)][k + 2*(lane>>4) + 1] }
// So each wave: 4 loads + 16 WMMAs per K-step, covering the full 4x4 grid of
// 16x16 output tiles. K is split into NWCHUNK wave-chunks per matrix; each
// wave writes a coalesced 64x64 fp32 partial (no atomics -> deterministic).
// ---------------------------------------------------------------------------
__global__ __launch_bounds__(128) void gram_wmma_kernel(
    const float* __restrict__ Ha, const float* __restrict__ Hs,
    float* __restrict__ Pa, float* __restrict__ Ps)
{
    const int mat  = blockIdx.x >> 6;          // 64 blocks per matrix
    const int blk  = blockIdx.x & 63;
    const int wave = threadIdx.x >> 5;         // 4 waves per block
    const int lane = threadIdx.x & 31;
    const int half = lane >> 4;
    const int l    = lane & 15;

    const float* __restrict__ X = mat ? Hs : Ha;
    float* __restrict__ P       = mat ? Ps : Pa;

    const int p  = blk * 4 + wave;             // wave-chunk id, 0..255
    const int k0 = p * KSUB;

    const float* rowp[4];
    #pragma unroll
    for (int t = 0; t < 4; ++t)
        rowp[t] = X + (size_t)(t * 16 + l) * K_FEAT + 2 * half;

    v8f acc[16];
    #pragma unroll
    for (int t = 0; t < 16; ++t) acc[t] = (v8f){0.f,0.f,0.f,0.f,0.f,0.f,0.f,0.f};

    for (int k = k0; k < k0 + KSUB; k += 4) {
        v2f f[4];
        #pragma unroll
        for (int t = 0; t < 4; ++t)
            f[t] = *(const v2f*)(rowp[t] + k);
        #pragma unroll
        for (int mi = 0; mi < 4; ++mi)
            #pragma unroll
            for (int ni = 0; ni < 4; ++ni)
                acc[mi * 4 + ni] = __builtin_amdgcn_wmma_f32_16x16x4_f32(
                    /*neg_a=*/false, f[mi],
                    /*neg_b=*/false, f[ni],
                    /*c_mod=*/(short)0, acc[mi * 4 + ni],
                    /*reuse_a=*/false, /*reuse_b=*/false);
    }

    // Coalesced partial store: element e = t*256 + lane*8 + v
    float* dst = P + (size_t)p * 4096;
    #pragma unroll
    for (int t = 0; t < 16; ++t)
        *(v8f*)(dst + t * 256 + lane * 8) = acc[t];
}

// ---------------------------------------------------------------------------
// Kernel 3: reduce NWCHUNK partials -> G (decoding the WMMA C/D layout:
// vgpr v, lane -> M = v + 8*(lane>>4), N = lane&15).
// ---------------------------------------------------------------------------
__global__ __launch_bounds__(256) void gram_reduce_kernel(
    const float* __restrict__ Pa, const float* __restrict__ Ps,
    float* __restrict__ Ga, float* __restrict__ Gs)
{
    const int gid = blockIdx.x * 256 + threadIdx.x;   // 0..8191
    const int mat = gid >> 12;
    const int e   = gid & 4095;
    const float* __restrict__ P = mat ? Ps : Pa;
    float* __restrict__ G       = mat ? Gs : Ga;

    float sum = 0.f;
    for (int p = 0; p < NWCHUNK; ++p)
        sum += P[(size_t)p * 4096 + e];

    const int t = e >> 8, lane = (e >> 3) & 31, v = e & 7;
    const int half = lane >> 4, l = lane & 15;
    const int mi = t >> 2, ni = t & 3;
    const int m = mi * 16 + v + 8 * half;
    const int n = ni * 16 + l;
    G[m * 64 + n] = sum;
}

// ---------------------------------------------------------------------------
// Kernel 4: cosine sims from Gram diag, -inf diagonal, row softmax, +eps,
// KL(p_sou || p_ada) / B. One block, 64 threads (one row each).
// ---------------------------------------------------------------------------
__global__ __launch_bounds__(64) void finalize_kernel(
    const float* __restrict__ Ga, const float* __restrict__ Gs,
    float* __restrict__ out)
{
    __shared__ float na[64], ns[64], klp[64];
    __shared__ float SA[64 * 64], SS[64 * 64];
    const int i = threadIdx.x;

    na[i] = fmaxf(sqrtf(fmaxf(Ga[i * 64 + i], 0.f)), EPS_COS);
    ns[i] = fmaxf(sqrtf(fmaxf(Gs[i * 64 + i], 0.f)), EPS_COS);
    __syncthreads();

    float maxA = -INFINITY, maxS = -INFINITY;
    for (int j = 0; j < 64; ++j) {
        float va = (j == i) ? -INFINITY : Ga[i * 64 + j] / (na[i] * na[j]);
        float vs = (j == i) ? -INFINITY : Gs[i * 64 + j] / (ns[i] * ns[j]);
        SA[i * 64 + j] = va;
        SS[i * 64 + j] = vs;
        maxA = fmaxf(maxA, va);
        maxS = fmaxf(maxS, vs);
    }
    float Za = 0.f, Zs = 0.f;
    for (int j = 0; j < 64; ++j) {
        float ea = (j == i) ? 0.f : expf(SA[i * 64 + j] - maxA);
        float es = (j == i) ? 0.f : expf(SS[i * 64 + j] - maxS);
        SA[i * 64 + j] = ea;
        SS[i * 64 + j] = es;
        Za += ea;
        Zs += es;
    }
    float acc = 0.f;
    for (int j = 0; j < 64; ++j) {
        float pa = SA[i * 64 + j] / Za + EPS_P;
        float ps = SS[i * 64 + j] / Zs + EPS_P;
        acc += ps * (logf(ps) - logf(pa));
    }
    klp[i] = acc;
    __syncthreads();
    if (i == 0) {
        float s = 0.f;
        for (int j = 0; j < 64; ++j) s += klp[j];
        out[0] = s / (float)BATCH;
    }
}

// ---------------------------------------------------------------------------
// Launch
// ---------------------------------------------------------------------------
extern "C" void kernel_launch(void* const* d_in, const int* in_sizes, int n_in,
                              void* d_out, int out_size, void* d_ws, size_t ws_size,
                              hipStream_t stream) {
    const float* z_ada = (const float*)d_in[0];
    const float* z_sou = (const float*)d_in[1];
    float* ws = (float*)d_ws;

    float* Ha = ws;                                      // 64 * 196608
    float* Hs = Ha + (size_t)BATCH * K_FEAT;             // 64 * 196608
    float* Pa = Hs + (size_t)BATCH * K_FEAT;             // 256 * 4096
    float* Ps = Pa + (size_t)NWCHUNK * 4096;             // 256 * 4096
    float* Ga = Ps + (size_t)NWCHUNK * 4096;             // 4096
    float* Gs = Ga + 4096;                               // 4096

    haar_norm_kernel<<<BATCH * CHANS, 256, 0, stream>>>(z_ada, Ha);
    haar_norm_kernel<<<BATCH * CHANS, 256, 0, stream>>>(z_sou, Hs);
    gram_wmma_kernel<<<128, 128, 0, stream>>>(Ha, Hs, Pa, Ps);
    gram_reduce_kernel<<<32, 256, 0, stream>>>(Pa, Ps, Ga, Gs);
    finalize_kernel<<<1, 64, 0, stream>>>(Ga, Gs, (float*)d_out);
}